// DenseMotionNetwork_29935922053141
// MI455X (gfx1250) — compile-verified
//
#include <hip/hip_runtime.h>
#include <cstdint>
#include <cstddef>

// Problem shape (fixed by the reference): input (8,32,16,64,64) f32,
// grid (8,16,64,64,3) f32, align_corners scalar int, out (8,32,16,64,64) f32.
#define NB    8
#define CCH   32
#define IDZ   16
#define IHY   64
#define IWX   64
#define PIN   (IDZ * IHY * IWX)     // 65536 voxels per (n,c) slice = 256 KB f32
#define POUT  (IDZ * IHY * IWX)     // D*H*W output points per batch = 65536
#define TPB   1024                  // 32 wave32s per workgroup
#define SPLIT 4                     // point-tiles per (n,c) -> 1024 workgroups
#define PSP   (POUT / SPLIT)        // 16384 points per workgroup

#if defined(__AMDGCN__) && __has_builtin(__builtin_amdgcn_global_load_async_to_lds_b128)
#define HAVE_ASYNC_LDS 1
#else
#define HAVE_ASYNC_LDS 0
#endif

// The async-to-LDS builtin takes (int4 AS1*, int4 AS3*, imm offset, imm cpol):
// clang diagnostic showed param0 as
//   '__attribute__((__vector_size__(4 * sizeof(int)))) int __device__ *'
typedef int v4i_t __attribute__((vector_size(16)));
typedef __attribute__((address_space(1))) v4i_t gbl_v4i_t;
typedef __attribute__((address_space(3))) v4i_t lds_v4i_t;

__device__ __forceinline__ int iclamp(int v, int lo, int hi) {
  v = (v < lo) ? lo : v;
  return (v > hi) ? hi : v;
}

__global__ __launch_bounds__(TPB, 1)
void grid_sample3d_lds_kernel(const float* __restrict__ inp,
                              const float* __restrict__ grid,
                              const int*   __restrict__ alignp,
                              float*       __restrict__ out)
{
  extern __shared__ float slice[];          // PIN floats = 256 KB dynamic LDS

  const int tid = (int)threadIdx.x;
  const unsigned wg = blockIdx.x;           // 0 .. NB*CCH*SPLIT-1
  const int s = (int)(wg % SPLIT);
  const int c = (int)((wg / SPLIT) % CCH);
  const int n = (int)(wg / (SPLIT * CCH));

  // ---------------- stage the (n,c) input slice into LDS ----------------
  const float* __restrict__ src = inp + (size_t)(n * CCH + c) * (size_t)PIN;

#if HAVE_ASYNC_LDS
  // 1024 threads x 16 iters x 16B = 256 KB, contiguous & coalesced.
#pragma unroll
  for (int j = 0; j < PIN / (TPB * 4); ++j) {
    const int e = (j * TPB + tid) * 4;      // float index, 16B granules
    __builtin_amdgcn_global_load_async_to_lds_b128(
        (gbl_v4i_t*)(src + e), (lds_v4i_t*)(slice + e), 0, 0);
  }
#if __has_builtin(__builtin_amdgcn_s_wait_asynccnt)
  __builtin_amdgcn_s_wait_asynccnt(0);
#else
  asm volatile("s_wait_asynccnt 0" ::: "memory");
#endif
#else
  // Fallback: global_load_b128 + ds_store_b128
#pragma unroll
  for (int j = 0; j < PIN / (TPB * 4); ++j) {
    const int e = (j * TPB + tid) * 4;
    *reinterpret_cast<float4*>(slice + e) =
        *reinterpret_cast<const float4*>(src + e);
  }
#endif
  __syncthreads();

  // ---------------- trilinear sampling out of LDS ----------------
  const int ac = *alignp;                   // uniform scalar
  const float* __restrict__ gbase = grid + (size_t)n * (size_t)POUT * 3u;
  float* __restrict__ obase = out + (size_t)(n * CCH + c) * (size_t)POUT;

  const int pbeg = s * PSP;
  const int pend = pbeg + PSP;

  for (int p = pbeg + tid; p < pend; p += TPB) {
    const float gx = gbase[3 * p + 0];
    const float gy = gbase[3 * p + 1];
    const float gz = gbase[3 * p + 2];

    float ix, iy, iz;
    if (ac) {
      ix = (gx + 1.0f) * 0.5f * (float)(IWX - 1);
      iy = (gy + 1.0f) * 0.5f * (float)(IHY - 1);
      iz = (gz + 1.0f) * 0.5f * (float)(IDZ - 1);
    } else {
      ix = ((gx + 1.0f) * (float)IWX - 1.0f) * 0.5f;
      iy = ((gy + 1.0f) * (float)IHY - 1.0f) * 0.5f;
      iz = ((gz + 1.0f) * (float)IDZ - 1.0f) * 0.5f;
    }

    const float fx0 = floorf(ix), fy0 = floorf(iy), fz0 = floorf(iz);
    const float fx = ix - fx0, fy = iy - fy0, fz = iz - fz0;
    const int x0 = (int)fx0, y0 = (int)fy0, z0 = (int)fz0;
    const int x1 = x0 + 1,   y1 = y0 + 1,   z1 = z0 + 1;

    const float wx0 = 1.0f - fx, wx1 = fx;
    const float wy0 = 1.0f - fy, wy1 = fy;
    const float wz0 = 1.0f - fz, wz1 = fz;

    // strict bounds masks on UNclamped corner indices (matches reference)
    const bool mx0 = (x0 > 0) & (x0 < IWX);
    const bool mx1 = (x1 > 0) & (x1 < IWX);
    const bool my0 = (y0 > 0) & (y0 < IHY);
    const bool my1 = (y1 > 0) & (y1 < IHY);
    const bool mz0 = (z0 > 0) & (z0 < IDZ);
    const bool mz1 = (z1 > 0) & (z1 < IDZ);

    // clamped indices for the gather
    const int x0c = iclamp(x0, 0, IWX - 1), x1c = iclamp(x1, 0, IWX - 1);
    const int y0c = iclamp(y0, 0, IHY - 1) * IWX;
    const int y1c = iclamp(y1, 0, IHY - 1) * IWX;
    const int z0c = iclamp(z0, 0, IDZ - 1) * (IWX * IHY);
    const int z1c = iclamp(z1, 0, IDZ - 1) * (IWX * IHY);

    auto term = [&](int xc, int ycs, int zcs, bool m, float w) -> float {
      const float v = slice[zcs + ycs + xc];   // ds_load_b32 gather
      return m ? v * w : 0.0f;
    };

    // accumulate in the reference's corner order
    float acc;
    acc  = term(x0c, y0c, z0c, mx0 && my0 && mz0, wx0 * wy0 * wz0); // tnw
    acc += term(x1c, y0c, z0c, mx1 && my0 && mz0, wx1 * wy0 * wz0); // tne
    acc += term(x0c, y1c, z0c, mx0 && my1 && mz0, wx0 * wy1 * wz0); // tsw
    acc += term(x1c, y1c, z0c, mx1 && my1 && mz0, wx1 * wy1 * wz0); // tse
    acc += term(x0c, y0c, z1c, mx0 && my0 && mz1, wx0 * wy0 * wz1); // bnw
    acc += term(x1c, y0c, z1c, mx1 && my0 && mz1, wx1 * wy0 * wz1); // bne
    acc += term(x0c, y1c, z1c, mx0 && my1 && mz1, wx0 * wy1 * wz1); // bsw
    acc += term(x1c, y1c, z1c, mx1 && my1 && mz1, wx1 * wy1 * wz1); // bse

    obase[p] = acc;                            // coalesced store
  }
}

extern "C" void kernel_launch(void* const* d_in, const int* in_sizes, int n_in,
                              void* d_out, int out_size, void* d_ws, size_t ws_size,
                              hipStream_t stream) {
  (void)in_sizes; (void)n_in; (void)out_size; (void)d_ws; (void)ws_size;
  const float* inp  = (const float*)d_in[0];
  const float* grd  = (const float*)d_in[1];
  const int*   ac   = (const int*)d_in[2];
  float*       out  = (float*)d_out;

  dim3 grid(NB * CCH * SPLIT);   // 1024 workgroups
  dim3 block(TPB);               // 1024 threads = 32 wave32s
  grid_sample3d_lds_kernel<<<grid, block, (size_t)PIN * sizeof(float), stream>>>(
      inp, grd, ac, out);
}